// BaseDenseAttention_18683107738258
// MI455X (gfx1250) — compile-verified
//
#include <hip/hip_runtime.h>

// CDNA5 (gfx1250) wave32 flash-attention:
//   - native FP32 WMMA (V_WMMA_F32_16X16X4_F32) for QK^T and P@V
//   - Tensor Data Mover (TENSOR_LOAD_TO_LDS) stages K/V tiles, double-buffered
//   - deterministic cross-wave score reduction through LDS

typedef float v2f __attribute__((ext_vector_type(2)));
typedef float v8f __attribute__((ext_vector_type(8)));
typedef unsigned int u32x4 __attribute__((ext_vector_type(4)));
typedef int i32x8 __attribute__((ext_vector_type(8)));
typedef int i32x4 __attribute__((ext_vector_type(4)));

#define T_SEQ   2048
#define DMODEL  1024
#define QTILE   16
#define KTILE   16
#define NWAVES  8
#define DSLICE  (DMODEL / NWAVES)     // 128 columns of D per wave
#define VROW    132                    // 128 data dwords + 4 pad dwords (TDM pad) per k-row
#define VTILE_FLOATS (KTILE * VROW)    // 2112 floats = 8448 B per tile buffer
#define VTILE_BYTES  (VTILE_FLOATS * 4)
#define SSTRIDE 17                     // padded row stride for score buffers

// One-instruction DMA of a 16(k) x 128(d) fp32 tile (row-major, row stride 1024)
// into LDS at lds_byte_addr with 4-dword padding after every 128 dwords
// (-> LDS row stride 132 floats, bank-conflict-free for both fragment patterns).
__device__ __forceinline__ void tdm_load_tile(unsigned lds_byte_addr, const float* gsrc)
{
    unsigned long long ga = (unsigned long long)(size_t)gsrc;
    u32x4 g0;
    g0.x = 1u;                                   // count = 1 valid descriptor
    g0.y = lds_byte_addr;                        // lds_addr (bits 63:32)
    g0.z = (unsigned)ga;                         // global_addr[31:0]
    g0.w = (unsigned)(ga >> 32) | (2u << 30);    // global_addr[56:32] | type=2 ("image")

    i32x8 g1;
    g1[0] = (int)((2u << 16)      // data_size = 4 bytes
                | (1u << 20)      // pad_enable
                | (6u << 22)      // pad_interval: pad after every 128 dwords
                | (3u << 25));    // pad_amount: 4 dwords
    g1[1] = (int)(1024u << 16);   // tensor_dim0 = 1024 (low 16 bits -> g1 bits 63:48)
    g1[2] = (int)(2048u << 16);   // tensor_dim1 = 2048 (low 16 bits -> g1 bits 95:80)
    g1[3] = (int)(128u  << 16);   // tile_dim0   = 128  (bits 127:112)
    g1[4] = 16;                   // tile_dim1   = 16   (bits 143:128), tile_dim2 = 0
    g1[5] = 1024;                 // tensor_dim0_stride = 1024 elements (bits 207:160)
    g1[6] = 0;
    g1[7] = 0;                    // tensor_dim1_stride = 0 (unused, 2D tile)

    i32x4 gz4 = {0, 0, 0, 0};     // groups 2/3 unused (2D)
    i32x8 gz8 = {0, 0, 0, 0, 0, 0, 0, 0};
    __builtin_amdgcn_tensor_load_to_lds(g0, g1, gz4, gz4, gz8, 0);
}

__global__ __launch_bounds__(256, 1)
void attn_causal_f32_wmma(const float* __restrict__ Q,
                          const float* __restrict__ V,
                          float* __restrict__ O)
{
    // Double-buffered per-wave V tiles, row-major [k][VROW]
    __shared__ float vbuf[NWAVES * 2][VTILE_FLOATS];   // 135168 B
    // Cross-wave score reduction buffers; sred[0] doubles as the P staging buffer.
    __shared__ float sred[NWAVES][QTILE * SSTRIDE];    //   8704 B

    const int tid  = threadIdx.x;
    const int wave = __builtin_amdgcn_readfirstlane(tid >> 5);  // wave-uniform -> SGPR
    const int lane = tid & 31;
    const int l16  = lane & 15;   // fragment column
    const int lh   = lane >> 4;   // half-wave select (+8 rows / +2 K)

    const int qt = blockIdx.x & 127;
    const int b  = blockIdx.x >> 7;
    const int q0 = qt * QTILE;
    const int d0 = wave * DSLICE;

    const float* Qb = Q + ((size_t)b * T_SEQ + q0) * DMODEL;
    const float* Vb = V + (size_t)b * T_SEQ * DMODEL;
    float*       Ob = O + ((size_t)b * T_SEQ + q0) * DMODEL;

    const unsigned lds_vbase = (unsigned)(size_t)(&vbuf[0][0]) + (unsigned)(wave * 2) * VTILE_BYTES;

    // ---- Load Q A-fragments once from global (L2-resident) ----
    // A-layout (16x4 chunk c): lane holds row M=l16, cols K = c*4 + 2*lh + {0,1}
    v2f aq[32];
#pragma unroll
    for (int c = 0; c < 32; ++c)
        aq[c] = *(const v2f*)(Qb + (size_t)l16 * DMODEL + d0 + c * 4 + 2 * lh);

    // ---- Online-softmax state ----
    v8f o_acc[8];
    v8f m, l;
#pragma unroll
    for (int j = 0; j < 8; ++j) o_acc[j] = v8f{};
#pragma unroll
    for (int v = 0; v < 8; ++v) { m[v] = -3.0e38f; l[v] = 0.0f; }

    // Prologue: start DMA of tile 0
    tdm_load_tile(lds_vbase, Vb + d0);

    for (int kt = 0; kt <= qt; ++kt) {
        __syncthreads();   // previous iteration's P-fragment reads of sred complete

        const float* vt = vbuf[wave * 2 + (kt & 1)];

        // Issue DMA for the next tile (double buffer), then wait for current tile.
        if (kt < qt) {
            asm volatile("s_wait_dscnt 0" ::: "memory");   // WAR: prior reads of target buffer done
            tdm_load_tile(lds_vbase + (unsigned)(((kt + 1) & 1) * VTILE_BYTES),
                          Vb + (size_t)(kt + 1) * KTILE * DMODEL + d0);
            __builtin_amdgcn_s_wait_tensorcnt((short)1);   // current tile landed
        } else {
            __builtin_amdgcn_s_wait_tensorcnt((short)0);
        }
        asm volatile("" ::: "memory");

        // ---- Partial scores: S_w = Q_slice(16x128) @ K_slice^T(128x16) ----
        // B-layout (4x16 chunk c): lane holds col N=l16, rows K = c*4 + 2*lh + {0,1}
        v8f s = v8f{};
#pragma unroll
        for (int c = 0; c < 32; ++c) {
            v2f bk = *(const v2f*)(vt + l16 * VROW + c * 4 + 2 * lh);
            s = __builtin_amdgcn_wmma_f32_16x16x4_f32(false, aq[c], false, bk,
                                                      (short)0, s, false, false);
        }

        // ---- Deterministic cross-wave reduction of partial scores ----
#pragma unroll
        for (int v = 0; v < 8; ++v)
            sred[wave][(v + 8 * lh) * SSTRIDE + l16] = s[v];
        __syncthreads();

        v8f sf;
#pragma unroll
        for (int v = 0; v < 8; ++v) {
            float acc = 0.0f;
#pragma unroll
            for (int w = 0; w < NWAVES; ++w)           // fixed order -> deterministic
                acc += sred[w][(v + 8 * lh) * SSTRIDE + l16];
            sf[v] = acc;
        }

        // ---- Causal mask on the diagonal tile (matches reference: s -= 1e9) ----
        if (kt == qt) {
#pragma unroll
            for (int v = 0; v < 8; ++v) {
                const int row = v + 8 * lh;
                sf[v] = (l16 > row) ? sf[v] - 1.0e9f : sf[v];
            }
        }

        // ---- Row max across 16 columns (within each 16-lane half) ----
        v8f tm = sf;
#pragma unroll
        for (int off = 1; off < 16; off <<= 1)
#pragma unroll
            for (int v = 0; v < 8; ++v)
                tm[v] = fmaxf(tm[v], __shfl_xor(tm[v], off, 32));

        v8f mnew, sc, p, rs;
#pragma unroll
        for (int v = 0; v < 8; ++v) {
            mnew[v] = fmaxf(m[v], tm[v]);
            sc[v]   = __expf(m[v] - mnew[v]);
            p[v]    = __expf(sf[v] - mnew[v]);
            rs[v]   = p[v];
        }
#pragma unroll
        for (int off = 1; off < 16; off <<= 1)
#pragma unroll
            for (int v = 0; v < 8; ++v)
                rs[v] += __shfl_xor(rs[v], off, 32);
#pragma unroll
        for (int v = 0; v < 8; ++v) {
            l[v] = l[v] * sc[v] + rs[v];
            m[v] = mnew[v];
        }
#pragma unroll
        for (int j = 0; j < 8; ++j)
#pragma unroll
            for (int v = 0; v < 8; ++v)
                o_acc[j][v] *= sc[v];

        // ---- Stage P (C-layout -> row-major) for A-fragment reload ----
        __syncthreads();                 // all waves finished reading sred
        if (wave == 0) {
#pragma unroll
            for (int v = 0; v < 8; ++v)
                sred[0][(v + 8 * lh) * SSTRIDE + l16] = p[v];
        }
        __syncthreads();

        v2f pa[4];
#pragma unroll
        for (int c = 0; c < 4; ++c) {
            const int col = c * 4 + 2 * lh;
            pa[c].x = sred[0][l16 * SSTRIDE + col];
            pa[c].y = sred[0][l16 * SSTRIDE + col + 1];
        }

        // ---- O(16x128) += P(16x16) @ V_tile(16x128) ----
#pragma unroll
        for (int j = 0; j < 8; ++j) {
#pragma unroll
            for (int c = 0; c < 4; ++c) {
                const int kk = c * 4 + 2 * lh;
                v2f bv;
                bv.x = vt[(kk + 0) * VROW + j * 16 + l16];
                bv.y = vt[(kk + 1) * VROW + j * 16 + l16];
                o_acc[j] = __builtin_amdgcn_wmma_f32_16x16x4_f32(false, pa[c], false, bv,
                                                                 (short)0, o_acc[j],
                                                                 false, false);
            }
        }
    }

    // ---- Normalize and store ----
    v8f linv;
#pragma unroll
    for (int v = 0; v < 8; ++v) linv[v] = 1.0f / l[v];
#pragma unroll
    for (int j = 0; j < 8; ++j)
#pragma unroll
        for (int v = 0; v < 8; ++v) {
            const int row = v + 8 * lh;
            Ob[(size_t)row * DMODEL + d0 + j * 16 + l16] = o_acc[j][v] * linv[v];
        }
}

extern "C" void kernel_launch(void* const* d_in, const int* in_sizes, int n_in,
                              void* d_out, int out_size, void* d_ws, size_t ws_size,
                              hipStream_t stream) {
    (void)n_in; (void)out_size; (void)d_ws; (void)ws_size;
    const float* Q = (const float*)d_in[0];
    const float* V = (const float*)d_in[1];
    float* O = (float*)d_out;

    const int B = in_sizes[0] / (T_SEQ * DMODEL);          // = 4
    dim3 grid(B * (T_SEQ / QTILE));                        // 512 workgroups
    dim3 block(NWAVES * 32);                               // 8 waves
    attn_causal_f32_wmma<<<grid, block, 0, stream>>>(Q, V, O);
}